// Net_91225105367814
// MI455X (gfx1250) — compile-verified
//
#include <hip/hip_runtime.h>
#include <math.h>

// G=256 graphs, N=512 points/graph, K=8 neighbors, H=16 features.
// Fused DGCNN-style kernel: one workgroup (512 threads = 16 wave32) per graph.
//  Phase 0: stage small weights into LDS.
//  Phase 1: encoder 4->16->16 (ELU), enc rows + squared norms into LDS.
//  Phase 2: Gram = enc @ enc^T via V_WMMA_F32_16X16X4_F32 (4 chained per 16x16
//           tile, K=16), column-blocks of 16 bounced through LDS; each thread
//           keeps a branch-free streaming top-8 (smallest d2) in registers.
//  Phase 3: EdgeConv 32->16 + ELU + max over 8 neighbors (VALU, xi-part hoisted).
//  Phase 4: mean pool per graph, head MLP 16->8->4->2, write outputs.

typedef float v2f __attribute__((ext_vector_type(2)));
typedef float v8f __attribute__((ext_vector_type(8)));

#define NG   256
#define NPT  512
#define KNN  8
#define HDIM 16

// LDS layout (floats). Row stride 18: conflict-free for the WMMA fragment
// gathers (lane L -> bank 18*L mod 64, all distinct for L=0..15, halves
// disjoint) and 8-byte aligned for paired loads.
#define LSTR      18
#define ENC_OFF   0                      // 512*18 = 9216
#define D2_OFF    9216                   // 512*18 = 9216 (reused as feat buf)
#define SQ_OFF    18432                  // 512
#define W_OFF     18944                  // 880 (W1 64|b1 16|W2 256|b2 16|Wc 512|bc 16)
#define POOL_OFF  19824                  // 16
#define SMEM_FLOATS 19840
#define SMEM_BYTES  (SMEM_FLOATS * 4)

__device__ __forceinline__ float elu1(float x) {
  // jax.nn.elu, alpha=1: x>0 ? x : expm1(x)
  return x > 0.0f ? x : expm1f(x);
}

__global__ __launch_bounds__(NPT) void dgcnn_fused_kernel(
    const float* __restrict__ x_pf,
    const float* __restrict__ W1,  const float* __restrict__ b1,
    const float* __restrict__ W2,  const float* __restrict__ b2,
    const float* __restrict__ Wc,  const float* __restrict__ bc,
    const float* __restrict__ Wo1, const float* __restrict__ bo1,
    const float* __restrict__ Wo2, const float* __restrict__ bo2,
    const float* __restrict__ Wo3, const float* __restrict__ bo3,
    float* __restrict__ out)
{
  extern __shared__ float smem[];
  float* encL  = smem + ENC_OFF;
  float* d2L   = smem + D2_OFF;
  float* sqL   = smem + SQ_OFF;
  float* wL    = smem + W_OFF;
  float* poolL = smem + POOL_OFF;

  const int g    = blockIdx.x;
  const int i    = threadIdx.x;      // point index within graph, 0..511
  const int lane = i & 31;
  const int wid  = i >> 5;           // wave id 0..15
  const int hsel = lane >> 4;        // half-wave select (0/1)
  const int lrow = lane & 15;

  // ---- Phase 0: weights -> LDS ----
  if (i < 64)  wL[0   + i] = W1[i];
  if (i < 16)  wL[64  + i] = b1[i];
  if (i < 256) wL[80  + i] = W2[i];
  if (i < 16)  wL[336 + i] = b2[i];
  wL[352 + i] = Wc[i];               // exactly 512 values
  if (i < 16)  wL[864 + i] = bc[i];
  __syncthreads();

  // ---- Phase 1: encoder ----
  const float4 xv = reinterpret_cast<const float4*>(x_pf)[g * NPT + i];
  float h1[HDIM];
  #pragma unroll
  for (int d = 0; d < HDIM; ++d) {
    float s = wL[64 + d]
            + xv.x * wL[0 * HDIM + d] + xv.y * wL[1 * HDIM + d]
            + xv.z * wL[2 * HDIM + d] + xv.w * wL[3 * HDIM + d];
    h1[d] = elu1(s);
  }
  float xi[HDIM];
  float sq = 0.0f;
  #pragma unroll
  for (int d = 0; d < HDIM; ++d) {
    float s = wL[336 + d];
    #pragma unroll
    for (int c = 0; c < HDIM; ++c) s += h1[c] * wL[80 + c * HDIM + d];
    s = elu1(s);
    xi[d] = s;
    sq += s * s;
    encL[i * LSTR + d] = s;
  }
  sqL[i] = sq;
  __syncthreads();

  // ---- Phase 2: Gram via WMMA + streaming top-8 ----
  float bd[KNN]; int bj[KNN];
  #pragma unroll
  for (int p = 0; p < KNN; ++p) { bd[p] = 3.0e38f; bj[p] = 0; }
  const float sq_i = sq;

  for (int jb = 0; jb < NPT / 16; ++jb) {
    const int j0 = jb << 4;
    // B fragment = enc^T columns j0..j0+15 (same gather pattern as A frags)
    v2f bf[4];
    #pragma unroll
    for (int s = 0; s < 4; ++s) {
      const float* p = encL + (j0 + lrow) * LSTR + (s * 4 + 2 * hsel);
      bf[s].x = p[0]; bf[s].y = p[1];
    }
    // 32 row-tiles over 16 waves -> 2 tiles per wave
    #pragma unroll
    for (int t = 0; t < 2; ++t) {
      const int m0 = ((wid << 1) | t) << 4;
      v2f af[4];
      #pragma unroll
      for (int s = 0; s < 4; ++s) {
        const float* p = encL + (m0 + lrow) * LSTR + (s * 4 + 2 * hsel);
        af[s].x = p[0]; af[s].y = p[1];
      }
      v8f acc = {};
      #pragma unroll
      for (int s = 0; s < 4; ++s)
        acc = __builtin_amdgcn_wmma_f32_16x16x4_f32(
            false, af[s], false, bf[s], (short)0, acc, false, false);
      // C/D layout: lane = column (lrow), VGPR v = row m0 + v + 8*hsel
      #pragma unroll
      for (int v = 0; v < 8; ++v)
        d2L[(m0 + v + (hsel << 3)) * LSTR + lrow] = acc[v];
    }
    __syncthreads();

    // each thread scans its row's 16 candidates (self included, like top_k)
    #pragma unroll
    for (int c = 0; c < 16; ++c) {
      const int j = j0 + c;
      float cd = sq_i + sqL[j] - 2.0f * d2L[i * LSTR + c];
      int   cj = j;
      // streaming sorted-insert, strict '<' keeps earlier-j on ties (top_k rule)
      #pragma unroll
      for (int p = 0; p < KNN; ++p) {
        const bool  lt = cd < bd[p];
        const float nd = lt ? cd : bd[p];
        const int   nj = lt ? cj : bj[p];
        const float od = lt ? bd[p] : cd;
        const int   oj = lt ? bj[p] : cj;
        bd[p] = nd; bj[p] = nj; cd = od; cj = oj;
      }
    }
    __syncthreads();
  }

  // ---- Phase 3: EdgeConv (32 -> 16, ELU, max over neighbors) ----
  const float* WcL = wL + 352;
  const float* bcL = wL + 864;
  float base[HDIM];
  #pragma unroll
  for (int d = 0; d < HDIM; ++d) {
    float s = bcL[d];
    #pragma unroll
    for (int c = 0; c < HDIM; ++c) s += xi[c] * WcL[c * HDIM + d];
    base[d] = s;
  }
  float feat[HDIM];
  #pragma unroll
  for (int d = 0; d < HDIM; ++d) feat[d] = -3.0e38f;
  #pragma unroll
  for (int n = 0; n < KNN; ++n) {
    const int j = bj[n];
    float dx[HDIM];
    #pragma unroll
    for (int c = 0; c < HDIM; ++c) dx[c] = encL[j * LSTR + c] - xi[c];
    #pragma unroll
    for (int d = 0; d < HDIM; ++d) {
      float s = base[d];
      #pragma unroll
      for (int c = 0; c < HDIM; ++c) s += dx[c] * WcL[(HDIM + c) * HDIM + d];
      feat[d] = fmaxf(feat[d], elu1(s));
    }
  }

  // ---- Phase 4: mean pool + head ----
  __syncthreads();                   // d2L free now; reuse as feat buffer
  #pragma unroll
  for (int d = 0; d < HDIM; ++d) d2L[i * LSTR + d] = feat[d];
  __syncthreads();
  if (i < HDIM) {
    float s = 0.0f;
    for (int r = 0; r < NPT; ++r) s += d2L[r * LSTR + i];
    poolL[i] = s * (1.0f / (float)NPT);
  }
  __syncthreads();
  if (i == 0) {
    float p[HDIM];
    #pragma unroll
    for (int c = 0; c < HDIM; ++c) p[c] = poolL[c];
    float o1[8];
    #pragma unroll
    for (int d = 0; d < 8; ++d) {
      float s = bo1[d];
      #pragma unroll
      for (int c = 0; c < HDIM; ++c) s += p[c] * Wo1[c * 8 + d];
      o1[d] = elu1(s);
    }
    float o2[4];
    #pragma unroll
    for (int d = 0; d < 4; ++d) {
      float s = bo2[d];
      #pragma unroll
      for (int c = 0; c < 8; ++c) s += o1[c] * Wo2[c * 4 + d];
      o2[d] = elu1(s);
    }
    #pragma unroll
    for (int d = 0; d < 2; ++d) {
      float s = bo3[d];
      #pragma unroll
      for (int c = 0; c < 4; ++c) s += o2[c] * Wo3[c * 2 + d];
      out[g * 2 + d] = s;
    }
    // second tuple element: batch = arange(G); store after the (G,2) outputs
    out[NG * 2 + g] = (float)g;
  }
}

extern "C" void kernel_launch(void* const* d_in, const int* in_sizes, int n_in,
                              void* d_out, int out_size, void* d_ws, size_t ws_size,
                              hipStream_t stream) {
  (void)in_sizes; (void)n_in; (void)out_size; (void)d_ws; (void)ws_size;
  const float* x_pf = (const float*)d_in[0];
  // d_in[1] = batch_pf (int32) is derivable from blockIdx; unused.
  const float* W1  = (const float*)d_in[2];
  const float* b1  = (const float*)d_in[3];
  const float* W2  = (const float*)d_in[4];
  const float* b2  = (const float*)d_in[5];
  const float* Wc  = (const float*)d_in[6];
  const float* bc  = (const float*)d_in[7];
  const float* Wo1 = (const float*)d_in[8];
  const float* bo1 = (const float*)d_in[9];
  const float* Wo2 = (const float*)d_in[10];
  const float* bo2 = (const float*)d_in[11];
  const float* Wo3 = (const float*)d_in[12];
  const float* bo3 = (const float*)d_in[13];
  float* out = (float*)d_out;

  dgcnn_fused_kernel<<<dim3(NG), dim3(NPT), SMEM_BYTES, stream>>>(
      x_pf, W1, b1, W2, b2, Wc, bc, Wo1, bo1, Wo2, bo2, Wo3, bo3, out);
}